// Multi_Grouping_19396072308836
// MI455X (gfx1250) — compile-verified
//
#include <hip/hip_runtime.h>
#include <hip/hip_bf16.h>

// ---------------- problem constants ----------------
#define BB 16
#define NN 4096
#define CH 64
#define SS 1024
#define KK 32
#define DD 128                       // 2*CH
#define MT (BB * SS * KK)            // 524288 total GEMM columns (power of two: 2^19)
#define LOG_MT 19
#define LOG_SK 15                    // S*K = 32768
#define CNT (SS * KK * CH)           // elements per batch for std
#define SLOPE 0.01f

typedef __attribute__((ext_vector_type(2))) float v2f;
typedef __attribute__((ext_vector_type(8))) float v8f;

// ---------------- 0: zero stat accumulators ----------------
__global__ void zero_stats_kernel(float* st) {
    int t = threadIdx.x;
    if (t < 528) st[t] = 0.0f;   // batch_sq[16] + sum1[128]+sq1[128]+sum2[128]+sq2[128]
}

// ---------------- 1: furthest point sampling (1 block / batch) ----------------
__global__ void fps_kernel(const float* __restrict__ xyz,
                           int* __restrict__ fps_idx,
                           float* __restrict__ out_xyz /* d_out, [B,S,3] */) {
    const int b = blockIdx.x;
    const int t = threadIdx.x;                 // 256 threads
    const float* xb = xyz + (size_t)b * NN * 3;

    __shared__ float bv[256];
    __shared__ int   bi2[256];
    __shared__ int   s_last;

    float md[NN / 256];
#pragma unroll
    for (int i = 0; i < NN / 256; ++i) md[i] = 1e10f;

    if (t == 0) {
        fps_idx[b * SS] = 0;
        out_xyz[(size_t)(b * SS) * 3 + 0] = xb[0];
        out_xyz[(size_t)(b * SS) * 3 + 1] = xb[1];
        out_xyz[(size_t)(b * SS) * 3 + 2] = xb[2];
        s_last = 0;
    }
    __syncthreads();
    int last = 0;

    for (int s = 1; s < SS; ++s) {
        const float lx = xb[last * 3 + 0];
        const float ly = xb[last * 3 + 1];
        const float lz = xb[last * 3 + 2];
        float bestv = -1.0f; int besti = 0;
#pragma unroll
        for (int i = 0; i < NN / 256; ++i) {
            const int n = t + i * 256;
            const float dx = xb[n * 3 + 0] - lx;
            const float dy = xb[n * 3 + 1] - ly;
            const float dz = xb[n * 3 + 2] - lz;
            const float d = dx * dx + dy * dy + dz * dz;
            md[i] = fminf(md[i], d);
            if (md[i] > bestv) { bestv = md[i]; besti = n; }
        }
        bv[t] = bestv; bi2[t] = besti;
        __syncthreads();
        for (int off = 128; off > 0; off >>= 1) {
            if (t < off) {
                const float v2 = bv[t + off]; const int i2 = bi2[t + off];
                if (v2 > bv[t] || (v2 == bv[t] && i2 < bi2[t])) { bv[t] = v2; bi2[t] = i2; }
            }
            __syncthreads();
        }
        if (t == 0) {
            const int n = bi2[0];
            s_last = n;
            fps_idx[b * SS + s] = n;
            out_xyz[(size_t)(b * SS + s) * 3 + 0] = xb[n * 3 + 0];
            out_xyz[(size_t)(b * SS + s) * 3 + 1] = xb[n * 3 + 1];
            out_xyz[(size_t)(b * SS + s) * 3 + 2] = xb[n * 3 + 2];
        }
        __syncthreads();
        last = s_last;
    }
}

// ---------------- 2: KNN (1 thread / query, LDS-tiled candidates) ----------------
__global__ void knn_kernel(const float* __restrict__ xyz,
                           const float* __restrict__ new_xyz,  // d_out front section
                           int* __restrict__ knn_idx) {
    const int q = blockIdx.x * blockDim.x + threadIdx.x;   // 0..B*S-1
    const int b = q >> 10;                                  // S = 1024
    const float* xb = xyz + (size_t)b * NN * 3;

    __shared__ float sx[512 * 3];

    const float qx = new_xyz[(size_t)q * 3 + 0];
    const float qy = new_xyz[(size_t)q * 3 + 1];
    const float qz = new_xyz[(size_t)q * 3 + 2];

    float bd[KK]; int bi[KK];
#pragma unroll
    for (int k = 0; k < KK; ++k) { bd[k] = 3.4e38f; bi[k] = 0; }
    float worst = 3.4e38f; int wpos = 0;

    for (int base = 0; base < NN; base += 512) {
        for (int j = threadIdx.x; j < 512 * 3; j += blockDim.x) sx[j] = xb[(size_t)base * 3 + j];
        __syncthreads();
        for (int n = 0; n < 512; ++n) {
            const float dx = sx[n * 3 + 0] - qx;
            const float dy = sx[n * 3 + 1] - qy;
            const float dz = sx[n * 3 + 2] - qz;
            const float d = dx * dx + dy * dy + dz * dz;
            if (d < worst) {
                bd[wpos] = d; bi[wpos] = base + n;
                worst = bd[0]; wpos = 0;
#pragma unroll
                for (int k = 1; k < KK; ++k)
                    if (bd[k] > worst) { worst = bd[k]; wpos = k; }
            }
        }
        __syncthreads();
    }
#pragma unroll
    for (int k = 0; k < KK; ++k) knn_idx[(size_t)q * KK + k] = bi[k];
}

// ---------------- 3: gather + mean-center + batch sumsq + concat new_pts ----------------
__global__ void group_a_kernel(const float* __restrict__ points,
                               const int* __restrict__ knn_idx,
                               const int* __restrict__ fps_idx,
                               float* __restrict__ feat,     // [128, MT]
                               float* __restrict__ batch_sq) {
    const int bs = blockIdx.x;
    const int b = bs >> 10;
    const int t = threadIdx.x;                  // 256 threads
    const float* pb = points + (size_t)b * NN * CH;
    const int* id = knn_idx + (size_t)bs * KK;

    __shared__ float g[KK * CH];                // 8 KB
    __shared__ float mean_sm[CH];

    for (int e = t; e < KK * CH; e += 256) {
        const int k = e >> 6, ch = e & 63;
        g[e] = pb[(size_t)id[k] * CH + ch];
    }
    __syncthreads();
    if (t < CH) {
        float m = 0.0f;
        for (int k = 0; k < KK; ++k) m += g[k * CH + t];
        mean_sm[t] = m * (1.0f / KK);
    }
    __syncthreads();

    float sq = 0.0f;
    for (int e = t; e < KK * CH; e += 256) {
        const int k = e >> 6, ch = e & 63;
        const float d = g[e] - mean_sm[ch];
        feat[(size_t)ch * MT + (size_t)bs * KK + k] = d;
        sq += d * d;
    }
    for (int m = 1; m < 32; m <<= 1) sq += __shfl_xor(sq, m, 32);
    if ((t & 31) == 0) atomicAdd(&batch_sq[b], sq);

    if (t < CH) {
        const float v = pb[(size_t)fps_idx[bs] * CH + t];
#pragma unroll
        for (int k = 0; k < KK; ++k)
            feat[(size_t)(CH + t) * MT + (size_t)bs * KK + k] = v;
    }
}

// ---------------- 4: affine normalize grouped channels (rows 0..63) ----------------
__global__ void group_b_kernel(float* __restrict__ feat,
                               const float* __restrict__ alpha,
                               const float* __restrict__ beta,
                               const float* __restrict__ batch_sq) {
    const size_t i = (size_t)blockIdx.x * 256 + threadIdx.x;   // 64*MT elements
    const int c = (int)(i >> LOG_MT);
    const size_t col = i & (size_t)(MT - 1);
    const int b = (int)(col >> LOG_SK);
    const float std = sqrtf(batch_sq[b] * (1.0f / (float)(CNT - 1)));
    feat[i] = alpha[c] * feat[i] / (std + 1e-5f) + beta[c];
}

// ---------------- 5: fp32 WMMA GEMM (Y = W @ X + bias) with BN stat epilogue ------
// Safe in place (Y == X): all reads precede the post-barrier writes per block.
__global__ void gemm_bn_stats_kernel(const float* __restrict__ X,
                                     float* __restrict__ Y,
                                     const float* __restrict__ W,
                                     const float* __restrict__ bias,
                                     float* __restrict__ gsum,
                                     float* __restrict__ gsq) {
    __shared__ float ls[DD], lq[DD];
    const int t = threadIdx.x;                    // 256 threads = 8 waves
    if (t < DD) { ls[t] = 0.0f; lq[t] = 0.0f; }

    const int lane = t & 31;
    const int wave = t >> 5;
    const int rb = wave * 16;                      // 16-row tile per wave
    const int cb0 = blockIdx.x * 64;               // 64-col stripe per block
    const int half = lane >> 4;                    // 0: K{0,1}  1: K{2,3}
    const int l16 = lane & 15;
    const int koff = half * 2;

    v8f acc[4] = {};
    const float* Wr = W + (size_t)(rb + l16) * DD;

    for (int kk = 0; kk < DD; kk += 4) {
        v2f a;
        a.x = Wr[kk + koff];
        a.y = Wr[kk + koff + 1];
#pragma unroll
        for (int j = 0; j < 4; ++j) {
            const int cb = cb0 + j * 16;
            const float* xc = X + (size_t)(kk + koff) * MT + cb + l16;
            v2f bf;
            bf.x = xc[0];
            bf.y = xc[MT];
            acc[j] = __builtin_amdgcn_wmma_f32_16x16x4_f32(
                false, a, false, bf, (short)0, acc[j], false, false);
        }
    }

    __syncthreads();   // all X reads done -> safe to write in place; orders ls/lq zeroing too

#pragma unroll
    for (int j = 0; j < 4; ++j) {
        const int cb = cb0 + j * 16;
#pragma unroll
        for (int r = 0; r < 8; ++r) {
            const int row = rb + r + half * 8;
            const float v = acc[j][r] + bias[row];
            Y[(size_t)row * MT + cb + l16] = v;
            float s = v, q = v * v;
#pragma unroll
            for (int m = 1; m < 16; m <<= 1) { s += __shfl_xor(s, m, 32); q += __shfl_xor(q, m, 32); }
            if (l16 == 0) { atomicAdd(&ls[row], s); atomicAdd(&lq[row], q); }
        }
    }
    __syncthreads();
    if (t < DD) { atomicAdd(&gsum[t], ls[t]); atomicAdd(&gsq[t], lq[t]); }
}

// ---------------- 6: finalize BN affine: y = A*h + Bc ----------------
__global__ void bn_finalize_kernel(const float* __restrict__ gsum,
                                   const float* __restrict__ gsq,
                                   const float* __restrict__ g,
                                   const float* __restrict__ be,
                                   float* __restrict__ A,
                                   float* __restrict__ Bc) {
    const int c = threadIdx.x;   // 128
    const float m = gsum[c] * (1.0f / (float)MT);
    const float var = gsq[c] * (1.0f / (float)MT) - m * m;
    const float a = g[c] * rsqrtf(var + 1e-5f);
    A[c] = a;
    Bc[c] = be[c] - m * a;
}

// ---------------- 7: elementwise BN-apply + leaky relu (in place) ----------------
__global__ void act_kernel(float* __restrict__ H,
                           const float* __restrict__ A,
                           const float* __restrict__ Bc) {
    const size_t i = (size_t)blockIdx.x * 256 + threadIdx.x;   // 128*MT elements
    const int c = (int)(i >> LOG_MT);
    const float v = A[c] * H[i] + Bc[c];
    H[i] = v > 0.0f ? v : SLOPE * v;
}

// ---------------- 8: BN2 + residual + leaky + max-pool over K ----------------
__global__ void final_pool_kernel(const float* __restrict__ H2,
                                  const float* __restrict__ Xf,
                                  const float* __restrict__ A,
                                  const float* __restrict__ Bc,
                                  float* __restrict__ out) {
    const int bs = blockIdx.x;
    const int c = threadIdx.x;   // 128
    const size_t base = (size_t)c * MT + (size_t)bs * KK;
    const float a = A[c], b = Bc[c];
    float mx = -3.4e38f;
#pragma unroll
    for (int k = 0; k < KK; ++k) {
        float v = a * H2[base + k] + b + Xf[base + k];
        v = v > 0.0f ? v : SLOPE * v;
        mx = fmaxf(mx, v);
    }
    out[(size_t)BB * SS * 3 + (size_t)bs * DD + c] = mx;
}

// ---------------- launcher ----------------
extern "C" void kernel_launch(void* const* d_in, const int* in_sizes, int n_in,
                              void* d_out, int out_size, void* d_ws, size_t ws_size,
                              hipStream_t stream) {
    const float* xyz    = (const float*)d_in[0];
    const float* points = (const float*)d_in[1];
    const float* alpha  = (const float*)d_in[2];
    const float* beta   = (const float*)d_in[3];
    const float* w1     = (const float*)d_in[4];
    const float* b1     = (const float*)d_in[5];
    const float* g1     = (const float*)d_in[6];
    const float* be1    = (const float*)d_in[7];
    const float* w2     = (const float*)d_in[8];
    const float* b2     = (const float*)d_in[9];
    const float* g2     = (const float*)d_in[10];
    const float* be2    = (const float*)d_in[11];
    float* out = (float*)d_out;

    float* ws = (float*)d_ws;
    float* Xf = ws;                               // [128, MT] feature (kept for residual)
    float* Hf = ws + (size_t)DD * MT;             // [128, MT] hidden (in-place gemm2)
    float* st = ws + (size_t)2 * DD * MT;
    float* batch_sq = st;                         // 16
    float* sum1 = st + 16;                        // 128
    float* sq1  = st + 144;
    float* sum2 = st + 272;
    float* sq2  = st + 400;
    float* A1   = st + 528;
    float* B1   = st + 656;
    float* A2   = st + 784;
    float* B2   = st + 912;
    int* fpsIdx = (int*)(st + 1040);              // B*S
    int* knnIdx = fpsIdx + BB * SS;               // B*S*K

    zero_stats_kernel<<<1, 1024, 0, stream>>>(st);
    fps_kernel<<<BB, 256, 0, stream>>>(xyz, fpsIdx, out);
    knn_kernel<<<(BB * SS) / 128, 128, 0, stream>>>(xyz, out, knnIdx);
    group_a_kernel<<<BB * SS, 256, 0, stream>>>(points, knnIdx, fpsIdx, Xf, batch_sq);
    group_b_kernel<<<(CH * MT) / 256, 256, 0, stream>>>(Xf, alpha, beta, batch_sq);

    gemm_bn_stats_kernel<<<MT / 64, 256, 0, stream>>>(Xf, Hf, w1, b1, sum1, sq1);
    bn_finalize_kernel<<<1, DD, 0, stream>>>(sum1, sq1, g1, be1, A1, B1);
    act_kernel<<<(DD * MT) / 256, 256, 0, stream>>>(Hf, A1, B1);

    gemm_bn_stats_kernel<<<MT / 64, 256, 0, stream>>>(Hf, Hf, w2, b2, sum2, sq2);
    bn_finalize_kernel<<<1, DD, 0, stream>>>(sum2, sq2, g2, be2, A2, B2);
    final_pool_kernel<<<BB * SS, DD, 0, stream>>>(Hf, Xf, A2, B2, out);
}